// GatingNetwork_3874060501222
// MI455X (gfx1250) — compile-verified
//
#include <hip/hip_runtime.h>
#include <math.h>

#define B_ 8
#define S_ 4096
#define D_ 2048
#define E_ 64
#define NCHUNK 64
#define ROWS_PER_CHUNK (S_ / NCHUNK)   // 64 rows per partial-sum chunk
#define BD (B_ * D_)                   // 16384 pooled elements

typedef __attribute__((ext_vector_type(2))) float v2f;
typedef __attribute__((ext_vector_type(4))) float v4f;
typedef __attribute__((ext_vector_type(8))) float v8f;

// ---------------------------------------------------------------------------
// Kernel 1: partial mean-pool. grid = B_*NCHUNK blocks, 256 threads.
// Each block sums 64 rows of [2048] for one batch with non-temporal 128-bit
// (b128, TH=NT) loads: x is 256 MB streamed once (> 192 MB L2), so keep it
// out of L2; the 4 MB of partials are stored with default RT hint so they
// stay L2-resident for the reduce kernel.
// partial layout: [NCHUNK][B_ * D_]
// ---------------------------------------------------------------------------
__global__ __launch_bounds__(256) void pool_partial_kernel(
    const float* __restrict__ x, float* __restrict__ partial) {
  const int blk = blockIdx.x;
  const int b = blk / NCHUNK;
  const int c = blk % NCHUNK;
  const int t = threadIdx.x;

  const float* base = x + ((size_t)b * S_ + (size_t)c * ROWS_PER_CHUNK) * D_;
  const int d0 = t * 4;          // first 128-bit slot
  const int d1 = d0 + 1024;      // second 128-bit slot (D = 2*1024)

  v4f acc0 = {0.f, 0.f, 0.f, 0.f};
  v4f acc1 = {0.f, 0.f, 0.f, 0.f};
  for (int s = 0; s < ROWS_PER_CHUNK; ++s) {
    const v4f* row = (const v4f*)(base + (size_t)s * D_);
    v4f va = __builtin_nontemporal_load(&row[d0 >> 2]);
    v4f vb = __builtin_nontemporal_load(&row[d1 >> 2]);
    acc0 += va;
    acc1 += vb;
  }
  v4f* outp = (v4f*)(partial + (size_t)c * BD + (size_t)b * D_);
  outp[d0 >> 2] = acc0;
  outp[d1 >> 2] = acc1;
}

// ---------------------------------------------------------------------------
// Kernel 2: fold NCHUNK partials -> x_mean [B_, D_], scale by 1/S.
// grid = BD/256 blocks, 256 threads; stride-BD reads are coalesced and hit L2.
// ---------------------------------------------------------------------------
__global__ __launch_bounds__(256) void pool_reduce_kernel(
    const float* __restrict__ partial, float* __restrict__ xmean) {
  const int idx = blockIdx.x * 256 + threadIdx.x;   // 0 .. BD-1
  float s = 0.f;
  #pragma unroll 4
  for (int c = 0; c < NCHUNK; ++c) s += partial[(size_t)c * BD + idx];
  xmean[idx] = s * (1.0f / (float)S_);
}

// ---------------------------------------------------------------------------
// Kernel 3: logits via V_WMMA_F32_16X16X4_F32, bias, top-2, softmax.
// One block, 128 threads = 4 waves; wave w handles experts [16w, 16w+16).
//
// A (16x4 f32): lanes 0-15 -> M=lane, v0=K0 v1=K1; lanes 16-31 -> M=lane-16,
//   v0=K2 v1=K3. Rows M>=8 are zeroed via a branchless lane mask (batch is 8).
// B (4x16 f32, mirrored): N = lane%16 (expert within tile), same K split.
// C/D (16x16 f32): VGPR j = row M=j (lanes 0-15) / M=j+8 (lanes 16-31).
// ---------------------------------------------------------------------------
__global__ __launch_bounds__(128) void gate_topk_kernel(
    const float* __restrict__ xmean, const float* __restrict__ W,
    const float* __restrict__ bias, float* __restrict__ out) {
  __shared__ float logitsLds[B_][E_];

  const int tid  = threadIdx.x;
  const int wave = tid >> 5;
  const int lane = tid & 31;
  const int laneLo = lane & 15;
  const int koff = (lane < 16) ? 0 : 2;

  const int etile = wave * 16;
  // Branchless A-row handling: invalid rows (M >= 8) alias a valid row and
  // get multiplied by 0, so the K-loop has no exec-mask divergence.
  const float amask = (laneLo < B_) ? 1.0f : 0.0f;
  const float* arow = xmean + (size_t)(laneLo & (B_ - 1)) * D_;
  const float* wrow = W + (size_t)(etile + laneLo) * D_;    // B col N=lane%16

  v8f acc = {0.f, 0.f, 0.f, 0.f, 0.f, 0.f, 0.f, 0.f};
  #pragma unroll 8
  for (int k = 0; k < D_; k += 4) {
    v2f a, bm;
    a.x = arow[k + koff] * amask;
    a.y = arow[k + koff + 1] * amask;
    bm.x = wrow[k + koff];
    bm.y = wrow[k + koff + 1];
    // 8 args: (neg_a, A, neg_b, B, c_mod, C, reuse_a, reuse_b)
    acc = __builtin_amdgcn_wmma_f32_16x16x4_f32(
        false, a, false, bm, (short)0, acc, false, false);
  }

  // Valid logits: acc[j], lanes 0-15 -> logits[b=j][e=etile+lane]
  if (lane < 16) {
    const float bb = bias[etile + lane];
    #pragma unroll
    for (int j = 0; j < B_; ++j)
      logitsLds[j][etile + lane] = acc[j] + bb;
  }
  __syncthreads();

  // Top-2 + softmax per batch row (8 lanes of wave 0 do the 64-wide scan).
  // Strict '>' matches jax.lax.top_k tie-breaking (earliest index wins).
  if (tid < B_) {
    const int b = tid;
    float v0 = -INFINITY, v1 = -INFINITY;
    int i0 = 0, i1 = 0;
    #pragma unroll
    for (int e = 0; e < E_; ++e) {
      const float v = logitsLds[b][e];
      if (v > v0)       { v1 = v0; i1 = i0; v0 = v; i0 = e; }
      else if (v > v1)  { v1 = v; i1 = e; }
    }
    // softmax over {v0, v1}; v0 is the max so exp(v0-v0)=1
    const float e1 = 1.0f;
    const float e2 = __expf(v1 - v0);
    const float inv = 1.0f / (e1 + e2);
    out[b * 2 + 0] = e1 * inv;
    out[b * 2 + 1] = e2 * inv;
    out[2 * B_ + b * 2 + 0] = (float)i0;   // indices, flattened after weights
    out[2 * B_ + b * 2 + 1] = (float)i1;
  }
}

// ---------------------------------------------------------------------------
extern "C" void kernel_launch(void* const* d_in, const int* in_sizes, int n_in,
                              void* d_out, int out_size, void* d_ws, size_t ws_size,
                              hipStream_t stream) {
  const float* x    = (const float*)d_in[0];   // [8, 4096, 2048]
  const float* W    = (const float*)d_in[1];   // [64, 2048]
  const float* bias = (const float*)d_in[2];   // [64]
  float* out = (float*)d_out;                  // 16 weights + 16 indices

  // Workspace: partial sums [NCHUNK][BD] (4 MB) then x_mean [BD] (64 KB)
  float* partial = (float*)d_ws;
  float* xmean   = partial + (size_t)NCHUNK * BD;

  pool_partial_kernel<<<dim3(B_ * NCHUNK), 256, 0, stream>>>(x, partial);
  pool_reduce_kernel<<<dim3(BD / 256), 256, 0, stream>>>(partial, xmean);
  gate_topk_kernel<<<dim3(1), 128, 0, stream>>>(xmean, W, bias, out);
}